// R2D2_4303557231318
// MI455X (gfx1250) — compile-verified
//
#include <hip/hip_runtime.h>

typedef __attribute__((ext_vector_type(2))) float v2f;
typedef __attribute__((ext_vector_type(8))) float v8f;

#define BATCH 32
#define SSUP  25          // support shots per batch
#define QQRY  75          // queries per batch
#define CDIM  640         // channels
#define NWAY  5
#define SDIM  32          // padded support dim (25 -> 32)
#define QDIM  80          // padded query dim (75 -> 80)
#define SPAD  644         // LDS column pad: 644 % 64 == 4 (dodge bank conflicts)
#define GW    (SDIM + NWAY + 1)  // augmented [G | Y] width = 38
#define PCH   128         // channels pooled per block (base is 16B aligned)

// ---------------------------------------------------------------------------
// Kernel 1: global average pool over the 5x5 spatial dims (25 contiguous
// floats per channel). This phase carries ~205 MB and bounds the runtime, so
// it must run at HBM speed: each block stages 128 channels (12.8 KB) with
// float4 (global_load_b128) fully-coalesced loads into LDS, then 128 threads
// segment-reduce (stride 25 is coprime with the 64 LDS banks -> no conflicts).
// ---------------------------------------------------------------------------
__global__ __launch_bounds__(256) void pool25_kernel(const float* __restrict__ x,
                                                     float* __restrict__ out) {
  __shared__ float4 buf4[PCH * 25 / 4];          // 12.8 KB, 16B aligned
  float* buf = (float*)buf4;

  const int chanBase = blockIdx.x * PCH;
  const float4* src = (const float4*)(x + (size_t)chanBase * 25); // 16B aligned
  for (int idx = threadIdx.x; idx < PCH * 25 / 4; idx += 256)
    buf4[idx] = src[idx];
  __syncthreads();

  const int t = threadIdx.x;
  if (t < PCH) {
    float s = 0.f;
#pragma unroll
    for (int j = 0; j < 25; ++j) s += buf[t * 25 + j];
    out[chanBase + t] = s * (1.0f / 25.0f);
  }
}

// ---------------------------------------------------------------------------
// Kernel 2: one block (8 waves) per batch element.
//   G  = S S^T + 50 I      via v_wmma_f32_16x16x4_f32  (2x2 tiles, K=640)
//   X  = G^{-1} Y           Gaussian elimination in LDS (SPD + ridge)
//   A  = Q S^T              via v_wmma_f32_16x16x4_f32  (5x2 tiles, K=640)
//   out = scale * (A X)
// ---------------------------------------------------------------------------
__global__ __launch_bounds__(256) void r2d2_solve_kernel(
    const float* __restrict__ Spool,   // (B, 25, 640)
    const float* __restrict__ Qpool,   // (B, 75, 640)
    const int*   __restrict__ sy,      // (B, 25)
    const float* __restrict__ scale,   // (1,)
    float*       __restrict__ outp)    // (B*75, 5)
{
  __shared__ float sS[SDIM][SPAD];     // padded support features  (~82 KB)
  __shared__ float sG[SDIM][GW];       // augmented [G | Y]        (~4.9 KB)
  __shared__ float sX[SDIM][NWAY];     // ridge solution
  __shared__ float sA[QDIM][SDIM + 1]; // Q S^T                    (~10.6 KB)

  const int b    = blockIdx.x;
  const int tid  = threadIdx.x;
  const int lane = tid & 31;
  const int wave = tid >> 5;           // 8 waves of 32 (wave32)

  // ---- stage S (zero-padded to 32 rows) into LDS -------------------------
  const float* Sb = Spool + (size_t)b * SSUP * CDIM;
  for (int idx = tid; idx < SDIM * CDIM; idx += 256) {
    const int r = idx / CDIM, c = idx % CDIM;
    sS[r][c] = (r < SSUP) ? Sb[r * CDIM + c] : 0.f;
  }
  __syncthreads();

  // ---- G = S S^T : 2x2 tiles of 16x16, waves 0..3 ------------------------
  // A 16x4 f32 layout: lanes 0-15 hold {K+0,K+1}, lanes 16-31 hold {K+2,K+3}
  // for row M = lane&15.  B 4x16: row K striped across 16 lanes per VGPR,
  // upper half-wave holds the upper K pair (B = S^T, so same index math).
  if (wave < 4) {
    const int tM = wave >> 1, tN = wave & 1;
    const int rA = tM * 16 + (lane & 15);
    const int rB = tN * 16 + (lane & 15);
    const int ko = (lane >> 4) * 2;
    v8f acc = {};
    for (int k = 0; k < CDIM; k += 4) {
      v2f a, bb;
      a.x  = sS[rA][k + ko];
      a.y  = sS[rA][k + ko + 1];
      bb.x = sS[rB][k + ko];
      bb.y = sS[rB][k + ko + 1];
      acc = __builtin_amdgcn_wmma_f32_16x16x4_f32(
          false, a, false, bb, (short)0, acc, false, false);
    }
    // C/D layout: VGPR r -> M = tM*16 + r + 8*(lane>=16), N = tN*16 + (lane&15)
    const int cN    = tN * 16 + (lane & 15);
    const int mBase = tM * 16 + ((lane >> 4) * 8);
#pragma unroll
    for (int r = 0; r < 8; ++r) sG[mBase + r][cN] = acc[r];
  }
  __syncthreads();

  // ---- ridge diagonal, padded identity, one-hot RHS ----------------------
  if (tid < SDIM) {
    const int i = tid;
    if (i < SSUP) sG[i][i] += 50.0f;   // L2_LAMBDA
    else          sG[i][i]  = 1.0f;    // padded rows -> e_i (solution = 0)
    const int label = (i < SSUP) ? sy[b * SSUP + i] : -1;
#pragma unroll
    for (int kk = 0; kk < NWAY; ++kk)
      sG[i][SDIM + kk] = (label == kk) ? 1.0f : 0.0f;
  }
  __syncthreads();

  // ---- Gaussian elimination (G is SPD + 50I: no pivoting needed) ---------
  for (int p = 0; p < SDIM - 1; ++p) {
    if (tid > p && tid < SDIM) {
      const float fac = sG[tid][p] / sG[p][p];
      for (int c = p; c < GW - 1; ++c)   // cols p..36
        sG[tid][c] -= fac * sG[p][c];
    }
    __syncthreads();
  }

  // ---- back substitution: X = G^{-1} Y (one thread per RHS column) ------
  if (tid < NWAY) {
    const int kk = tid;
    for (int i = SDIM - 1; i >= 0; --i) {
      float v = sG[i][SDIM + kk];
      for (int j = i + 1; j < SDIM; ++j) v -= sG[i][j] * sX[j][kk];
      sX[i][kk] = v / sG[i][i];
    }
  }

  // ---- A = Q S^T : 5x2 tiles across 8 waves (needs only sS) --------------
  const float* Qb = Qpool + (size_t)b * QQRY * CDIM;
  for (int t = wave; t < 10; t += 8) {
    const int tM = t >> 1, tN = t & 1;
    const int rQ = tM * 16 + (lane & 15);
    const int rS = tN * 16 + (lane & 15);
    const int ko = (lane >> 4) * 2;
    const bool qv = rQ < QQRY;
    const float* qrow = Qb + (size_t)(qv ? rQ : 0) * CDIM;
    v8f acc = {};
    for (int k = 0; k < CDIM; k += 4) {
      v2f a, bb;
      a.x  = qv ? qrow[k + ko]     : 0.f;
      a.y  = qv ? qrow[k + ko + 1] : 0.f;
      bb.x = sS[rS][k + ko];
      bb.y = sS[rS][k + ko + 1];
      acc = __builtin_amdgcn_wmma_f32_16x16x4_f32(
          false, a, false, bb, (short)0, acc, false, false);
    }
    const int cN    = tN * 16 + (lane & 15);
    const int mBase = tM * 16 + ((lane >> 4) * 8);
#pragma unroll
    for (int r = 0; r < 8; ++r) sA[mBase + r][cN] = acc[r];
  }
  __syncthreads();

  // ---- logits = scale * (A X) -------------------------------------------
  const float sc = scale[0];
  for (int idx = tid; idx < QQRY * NWAY; idx += 256) {
    const int qi = idx / NWAY, kk = idx % NWAY;
    float v = 0.f;
#pragma unroll
    for (int j = 0; j < SSUP; ++j) v += sA[qi][j] * sX[j][kk];
    outp[((size_t)b * QQRY + qi) * NWAY + kk] = sc * v;
  }
}

// ---------------------------------------------------------------------------
extern "C" void kernel_launch(void* const* d_in, const int* in_sizes, int n_in,
                              void* d_out, int out_size, void* d_ws, size_t ws_size,
                              hipStream_t stream) {
  const float* support_xf = (const float*)d_in[0]; // (32,25,640,5,5)
  const int*   support_y  = (const int*)  d_in[1]; // (32,25)
  const float* query_xf   = (const float*)d_in[2]; // (32,75,640,5,5)
  // d_in[3] = query_y (unused), d_in[5]/d_in[6] = n_way/k_shot (hardcoded 5)
  const float* scale      = (const float*)d_in[4]; // (1,)
  float* out = (float*)d_out;

  float* Spool = (float*)d_ws;                         // 512000 floats
  float* Qpool = Spool + (size_t)BATCH * SSUP * CDIM;  // 1536000 floats

  const int supChans = BATCH * SSUP * CDIM;  // 512000  (divisible by 128)
  const int qryChans = BATCH * QQRY * CDIM;  // 1536000 (divisible by 128)

  pool25_kernel<<<supChans / PCH, 256, 0, stream>>>(support_xf, Spool);
  pool25_kernel<<<qryChans / PCH, 256, 0, stream>>>(query_xf,  Qpool);
  r2d2_solve_kernel<<<BATCH, 256, 0, stream>>>(Spool, Qpool, support_y, scale, out);
}